// MultiScalePeriodicPatchEmbedding_23622320128513
// MI455X (gfx1250) — compile-verified
//
#include <hip/hip_runtime.h>
#include <math.h>

typedef float v2f __attribute__((ext_vector_type(2)));
typedef float v8f __attribute__((ext_vector_type(8)));

#define SEQ   512
#define NC    8
#define DM    512
#define NB    8
#define PS    43
#define SUMN  1657
#define XCAT  (NB * NC * SUMN * DM)   // 54,296,576 floats
#define TOTAL_UNITS (32 * SUMN)       // 53,024 wave-units (16M x 64D tiles)

#define PERIOD_LIST 512,256,170,128,102,85,73,64,56,51,46,42,39,36,34,32,30,28, \
                    26,25,24,23,22,21,20,19,18,17,16,15,14,13,12,11,10,9,8,7,6,5,4,3,2
#define NPATCH_LIST 1,2,4,4,6,7,8,8,10,11,12,13,14,15,16,16,18,19,20,21,22,23,24, \
                    25,26,27,29,31,32,35,37,40,43,47,52,57,64,74,86,103,128,171,256
#define OFF_LIST    0,1,3,7,11,17,24,32,40,50,61,73,86,100,115,131,147,165,184,204, \
                    225,247,270,294,319,345,372,401,432,464,499,536,576,619,666,718, \
                    775,839,913,999,1102,1230,1401,1657

__constant__ int c_period[PS]  = { PERIOD_LIST };
__constant__ int c_N[PS]       = { NPATCH_LIST };
__constant__ int c_off[PS + 1] = { OFF_LIST };

static const int h_period[PS]  = { PERIOD_LIST };

struct WTab { const float* w[PS]; };

// ---------------------------------------------------------------------------
// Kernel 1: AFNO-style gating. One block. s -> DFT power -> MLP -> top-5 ->
// softmax. Writes gates + sel (float 0/1) to tail of d_out.
// ---------------------------------------------------------------------------
__global__ __launch_bounds__(256) void gate_kernel(
    const float* __restrict__ x,  const float* __restrict__ sw,
    const float* __restrict__ sb, const float* __restrict__ w1,
    const float* __restrict__ b1, const float* __restrict__ w2,
    const float* __restrict__ b2, const float* __restrict__ wg,
    float* __restrict__ gates_out, float* __restrict__ sel_out)
{
    __shared__ float sm[15360];           // 60 KB, regions reused across phases
    float* S   = sm;                      // [8*512]  start-linear series
    float* TWC = sm + 4096;               // [512]    cos twiddles
    float* TWS = sm + 4608;               // [512]    sin twiddles
    float* PW  = sm + 5120;               // [8*256]  power spectrum
    float* H   = sm + 7168;               // [8*1024] hidden
    float* Z   = sm;                      // [8*256]  reuses S (dead)
    float* LG  = sm + 2048;               // [8*43]   logits

    const int t = threadIdx.x;

    // ---- phase 1: s[b,l] = x[b,l,:] @ start_w + start_b ; twiddle table ----
    float bconst = sb[0];
    float w0c[8];
#pragma unroll
    for (int c = 0; c < 8; ++c) w0c[c] = sw[c];
    for (int idx = t; idx < NB * SEQ; idx += 256) {
        int b = idx >> 9, l = idx & 511;
        const float* xr = x + (b * SEQ + l) * NC;
        float acc = bconst;
#pragma unroll
        for (int c = 0; c < 8; ++c) acc += xr[c] * w0c[c];
        S[idx] = acc;
    }
    for (int idx = t; idx < 512; idx += 256) {
        float ang = -6.28318530717958647692f * (float)idx * (1.0f / 512.0f);
        TWC[idx] = cosf(ang);
        TWS[idx] = sinf(ang);
    }
    __syncthreads();

    // ---- phase 2: power[b,f-1] = |rfft(s)[f]|^2 / L (ortho), f = 1..256 ----
    for (int idx = t; idx < NB * 256; idx += 256) {
        int b = idx >> 8, f = (idx & 255) + 1;
        const float* srow = S + b * SEQ;
        float re = 0.f, im = 0.f;
        int m = 0;                         // exact phase index (f*l mod 512)
        for (int l = 0; l < SEQ; ++l) {
            float sv = srow[l];
            re += sv * TWC[m];
            im += sv * TWS[m];
            m = (m + f) & 511;
        }
        PW[idx] = (re * re + im * im) * (1.0f / 512.0f);
    }
    __syncthreads();

    // ---- phase 3: h = gelu(power @ w1 + b1), exact erf gelu ----
    for (int idx = t; idx < NB * 1024; idx += 256) {
        int b = idx >> 10, j = idx & 1023;
        const float* pwb = PW + b * 256;
        float a = b1[j];
        for (int f = 0; f < 256; ++f) a += pwb[f] * w1[f * 1024 + j];
        H[idx] = 0.5f * a * (1.0f + erff(a * 0.70710678118654752440f));
    }
    __syncthreads();

    // ---- phase 4: z = h @ w2 + b2 ----
    for (int idx = t; idx < NB * 256; idx += 256) {
        int b = idx >> 8, j = idx & 255;
        const float* hb = H + b * 1024;
        float a = b2[j];
        for (int k = 0; k < 1024; ++k) a += hb[k] * w2[k * 256 + j];
        Z[idx] = a;
    }
    __syncthreads();

    // ---- phase 5: logits = z @ w_gate ----
    if (t < NB * PS) {
        int b = t / PS, i = t - b * PS;
        const float* zb = Z + b * 256;
        float a = 0.f;
        for (int f = 0; f < 256; ++f) a += zb[f] * wg[f * PS + i];
        LG[b * PS + i] = a;
    }
    __syncthreads();

    // ---- phase 6: per-batch top-5 -> masked softmax -> gates / sel ----
    if (t < NB) {
        int b = t;
        float lg[PS];
        bool  ch[PS];
        for (int i = 0; i < PS; ++i) { lg[i] = LG[b * PS + i]; ch[i] = false; }
        float maxv = -3.4e38f;
        for (int k = 0; k < 5; ++k) {      // strict > keeps first index on ties
            float best = -3.4e38f; int bi = 0;
            for (int i = 0; i < PS; ++i)
                if (!ch[i] && lg[i] > best) { best = lg[i]; bi = i; }
            ch[bi] = true;
            if (k == 0) maxv = best;
        }
        float e[PS];
        float ssum = 0.f;
        for (int i = 0; i < PS; ++i) {
            e[i] = ch[i] ? expf(lg[i] - maxv) : 0.f;
            ssum += e[i];
        }
        for (int i = 0; i < PS; ++i) {
            float g = e[i] / ssum;
            gates_out[b * PS + i] = g;
            sel_out[b * PS + i]   = (g > 0.f) ? 1.f : 0.f;
        }
    }
}

// ---------------------------------------------------------------------------
// Kernel 2: per-scale patch embedding via f32 WMMA.
// One wave computes a 16(M) x 64(D) tile: A = edge-padded patches [64N, p],
// B = value_ws[i] [p, 512]. K stepped by 4 with V_WMMA_F32_16X16X4_F32.
// Main K-loop is branch-free (unconditional loads, clamped A addresses,
// pointer-increment B); a single masked epilogue step handles p mod 4.
// ---------------------------------------------------------------------------
__global__ __launch_bounds__(256) void embed_kernel(
    WTab wt, const float* __restrict__ x, const float* __restrict__ pt,
    const float* __restrict__ pc, const float* __restrict__ sel,
    float* __restrict__ out)
{
    const int unit = (blockIdx.x * 256 + threadIdx.x) >> 5;  // grid covers exactly
    const int lane = threadIdx.x & 31;
    const int half = lane >> 4, lm = lane & 15;

    // unit -> period: each period owns 32*N consecutive units
    const int q = unit >> 5;
    int i = 0;
    while (q >= c_off[i + 1]) ++i;
    const int p = c_period[i], N = c_N[i], off = c_off[i];
    const int rem   = unit - (c_off[i] << 5);
    const int mtile = rem >> 3;            // 0 .. 4N-1
    const int dbase = (rem & 7) << 6;      // 0,64,...,448

    // A-frag row for this lane: global row m = (b*8+c)*N + n
    const int mA = mtile * 16 + lm;
    const int bc = mA / N;
    const int nA = mA - bc * N;
    const int bA = bc >> 3, cA = bc & 7;
    const float* __restrict__ xr = x + bA * (SEQ * NC) + cA;  // x[b,l,c], stride NC
    const int l0 = nA * p;
    const float* __restrict__ W = wt.w[i];

    v8f acc0 = {}, acc1 = {}, acc2 = {}, acc3 = {};

    // ISA A/B layout: lanes 0-15 hold K = {k, k+1}, lanes 16-31 K = {k+2, k+3}
    int aoff = l0 + (half << 1);                      // series index of a.x
    const float* __restrict__ pB = W + (half << 1) * DM + dbase + lm;

    const int nfull = p >> 2;                         // branch-free main loop
    for (int ks = 0; ks < nfull; ++ks) {
        int la0 = aoff;     if (la0 > SEQ - 1) la0 = SEQ - 1;   // edge padding
        int la1 = aoff + 1; if (la1 > SEQ - 1) la1 = SEQ - 1;
        v2f a;
        a.x = xr[la0 * NC];
        a.y = xr[la1 * NC];
        v2f b0, b1_, b2_, b3;
        b0.x  = pB[0];        b0.y  = pB[DM];
        b1_.x = pB[16];       b1_.y = pB[DM + 16];
        b2_.x = pB[32];       b2_.y = pB[DM + 32];
        b3.x  = pB[48];       b3.y  = pB[DM + 48];
        acc0 = __builtin_amdgcn_wmma_f32_16x16x4_f32(false, a, false, b0,  (short)0, acc0, false, false);
        acc1 = __builtin_amdgcn_wmma_f32_16x16x4_f32(false, a, false, b1_, (short)0, acc1, false, false);
        acc2 = __builtin_amdgcn_wmma_f32_16x16x4_f32(false, a, false, b2_, (short)0, acc2, false, false);
        acc3 = __builtin_amdgcn_wmma_f32_16x16x4_f32(false, a, false, b3,  (short)0, acc3, false, false);
        aoff += 4;
        pB   += 4 * DM;
    }

    if (p & 3) {                                      // masked epilogue step
        const int kx = (p & ~3) + (half << 1);
        const float m0 = (kx     < p) ? 1.f : 0.f;    // mask A only: a*b == 0
        const float m1 = (kx + 1 < p) ? 1.f : 0.f;
        int la0 = aoff;     if (la0 > SEQ - 1) la0 = SEQ - 1;
        int la1 = aoff + 1; if (la1 > SEQ - 1) la1 = SEQ - 1;
        v2f a;
        a.x = xr[la0 * NC] * m0;
        a.y = xr[la1 * NC] * m1;
        const int rk0 = (kx     < p) ? kx     : p - 1;  // clamp rows: loads stay
        const int rk1 = (kx + 1 < p) ? kx + 1 : p - 1;  // in-bounds and finite
        const float* B0 = W + rk0 * DM + dbase + lm;
        const float* B1 = W + rk1 * DM + dbase + lm;
        v2f b0, b1_, b2_, b3;
        b0.x  = B0[0];   b0.y  = B1[0];
        b1_.x = B0[16];  b1_.y = B1[16];
        b2_.x = B0[32];  b2_.y = B1[32];
        b3.x  = B0[48];  b3.y  = B1[48];
        acc0 = __builtin_amdgcn_wmma_f32_16x16x4_f32(false, a, false, b0,  (short)0, acc0, false, false);
        acc1 = __builtin_amdgcn_wmma_f32_16x16x4_f32(false, a, false, b1_, (short)0, acc1, false, false);
        acc2 = __builtin_amdgcn_wmma_f32_16x16x4_f32(false, a, false, b2_, (short)0, acc2, false, false);
        acc3 = __builtin_amdgcn_wmma_f32_16x16x4_f32(false, a, false, b3,  (short)0, acc3, false, false);
    }

    // C/D layout: VGPR j = rows {mtile*16+j (lanes 0-15), mtile*16+8+j (16-31)}
    int   rowbase[8], ptn[8], pcc[8];
    float selv[8];
#pragma unroll
    for (int j = 0; j < 8; ++j) {
        const int row = mtile * 16 + j + (half << 3);
        const int rbc = row / N;
        const int n   = row - rbc * N;
        const int b   = rbc >> 3, c = rbc & 7;
        rowbase[j] = ((b * NC + c) * SUMN + off + n) * DM;
        ptn[j]     = n * DM;
        pcc[j]     = c * DM;
        selv[j]    = sel[b * PS + i];
    }
    const v8f accs[4] = { acc0, acc1, acc2, acc3 };
#pragma unroll
    for (int tt = 0; tt < 4; ++tt) {
        const int d = dbase + tt * 16 + lm;
#pragma unroll
        for (int j = 0; j < 8; ++j) {
            float v = (accs[tt][j] + pt[ptn[j] + d] + pc[pcc[j] + d]) * selv[j];
            __builtin_nontemporal_store(v, out + rowbase[j] + d);
        }
    }
}

// ---------------------------------------------------------------------------
extern "C" void kernel_launch(void* const* d_in, const int* in_sizes, int n_in,
                              void* d_out, int out_size, void* d_ws, size_t ws_size,
                              hipStream_t stream)
{
    (void)in_sizes; (void)out_size; (void)d_ws; (void)ws_size;

    const float* x   = (const float*)d_in[0];
    const float* sw  = (const float*)d_in[1];
    const float* sb  = (const float*)d_in[2];
    const float* w1  = (const float*)d_in[3];
    const float* b1  = (const float*)d_in[4];
    const float* w2  = (const float*)d_in[5];
    const float* b2  = (const float*)d_in[6];
    const float* wg  = (const float*)d_in[7];
    // d_in[8] = w_noise (unused by reference in eval mode)

    WTab wt;
    const float *pt, *pc;
    if (n_in >= 9 + PS + 2) {              // value_ws flattened to 43 inputs
        for (int i = 0; i < PS; ++i) wt.w[i] = (const float*)d_in[9 + i];
        pt = (const float*)d_in[9 + PS];
        pc = (const float*)d_in[9 + PS + 1];
    } else {                               // value_ws concatenated in one buffer
        const float* base = (const float*)d_in[9];
        long o = 0;
        for (int i = 0; i < PS; ++i) { wt.w[i] = base + o; o += (long)h_period[i] * DM; }
        pt = (const float*)d_in[10];
        pc = (const float*)d_in[11];
    }

    float* out   = (float*)d_out;
    float* gates = out + (size_t)XCAT;     // outputs concatenated: xcat, gates, sel
    float* selo  = gates + NB * PS;

    gate_kernel<<<1, 256, 0, stream>>>(x, sw, sb, w1, b1, w2, b2, wg, gates, selo);
    // 53,024 wave-units, 8 waves/block -> exactly 6628 blocks (no partial waves,
    // EXEC all-ones for WMMA)
    embed_kernel<<<TOTAL_UNITS / 8, 256, 0, stream>>>(wt, x, pt, pc, selo, out);
}